// Yolo2DObjDetLoss_63831803953669
// MI455X (gfx1250) — compile-verified
//
#include <hip/hip_runtime.h>
#include <hip/hip_bf16.h>
#include <stdint.h>

// Problem constants (from reference)
#define B_            8
#define NC_           6
#define CAMS          48          // B_*NC_
#define A_            25200
#define C_            10
#define MAX_GT        32
#define LAM_COORD     5.0f
#define LAM_NOOBJ     0.5f
#define LAM_OBJ       1.0f
#define LAM_CLS       1.0f
#define IOU_THR       0.5f

#define BLOCKS_PER_CAM 24
#define TPB            256

typedef float v2f  __attribute__((ext_vector_type(2)));
typedef float v8f  __attribute__((ext_vector_type(8)));
typedef unsigned int u32x4 __attribute__((ext_vector_type(4)));
typedef int   i32x8 __attribute__((ext_vector_type(8)));
typedef int   i32x4 __attribute__((ext_vector_type(4)));

// ---------------------------------------------------------------------------
// Kernel 0: compute per-camera [start, ngt] from start_end_idx (B x NC cumsum
// layout per reference) and zero the 48x6 accumulator block. Re-run on every
// launch (harness does not re-zero workspace between replays).
// ---------------------------------------------------------------------------
__global__ void yolo_setup_kernel(const int* __restrict__ se,
                                  int* __restrict__ starts,
                                  int* __restrict__ ngt,
                                  float* __restrict__ accs) {
  const int t = threadIdx.x;
  if (t == 0) {
    int off = 0;
    for (int b = 0; b < B_; ++b) {
      int prev = 0;
      for (int c = 0; c < NC_; ++c) {
        const int cur = se[b * NC_ + c];
        starts[b * NC_ + c] = off + prev;
        ngt[b * NC_ + c]    = cur - prev;
        prev = cur;
      }
      off += prev;   // offsets = exclusive cumsum of s[:, -1]
    }
  }
  for (int i = t; i < CAMS * 6; i += blockDim.x) accs[i] = 0.0f;
}

// ---------------------------------------------------------------------------
// Helpers
// ---------------------------------------------------------------------------
__device__ __forceinline__ float softplus_f(float x) {
  // numerically stable: max(x,0) + log1p(exp(-|x|))
  return fmaxf(x, 0.0f) + log1pf(expf(-fabsf(x)));
}

// Wave32 sum reduction on the matrix pipe.
// A (16x4 f32): VGPR0 lane m (0-15) -> A[m,0]; lane m+16 -> A[m,2]; VGPR1 = 0.
// B = all ones -> D[m,n] = v[lane m] + v[lane m+16] for every n.
// Sum the 8 D VGPRs per lane (covers M=0..7 in lanes 0-15, M=8..15 in lanes
// 16-31), then one cross-half shuffle-add gives the full 32-lane sum.
__device__ __forceinline__ float wave_sum_wmma(float v) {
  v2f a; a[0] = v;    a[1] = 0.0f;
  v2f b; b[0] = 1.0f; b[1] = 1.0f;
  v8f c = {};
  c = __builtin_amdgcn_wmma_f32_16x16x4_f32(false, a, false, b,
                                            (short)0, c, false, false);
  float s = c[0] + c[1] + c[2] + c[3] + c[4] + c[5] + c[6] + c[7];
  s += __shfl_xor(s, 16, 32);
  return s;
}

// ---------------------------------------------------------------------------
// Kernel 1: main streaming pass. Grid = CAMS * BLOCKS_PER_CAM blocks.
// GT boxes for the block's camera are DMA'd into LDS by the Tensor Data
// Mover (1-D descriptor, data_size=4B, tile_dim0 = ngt*4 elements).
// ---------------------------------------------------------------------------
__global__ void __launch_bounds__(TPB) yolo_loss_main(
    const float* __restrict__ pred_boxes,   // [CAMS, A, 4]
    const float* __restrict__ pred_labels,  // [CAMS, A, C]
    const float* __restrict__ pred_obj,     // [CAMS, A, 1]
    const float* __restrict__ gt_boxes,     // [CAMS*G, 4]
    const int*   __restrict__ gt_labels,    // [CAMS*G]
    const int*   __restrict__ starts,       // [CAMS]
    const int*   __restrict__ ngt_arr,      // [CAMS]
    float* __restrict__ accs)               // [CAMS][6]
{
  __shared__ float lds_gtb[MAX_GT * 4];
  __shared__ int   lds_gtl[MAX_GT];

  const int cam      = blockIdx.x / BLOCKS_PER_CAM;
  const int blkInCam = blockIdx.x % BLOCKS_PER_CAM;
  const int tid      = threadIdx.x;
  const int ngt      = ngt_arr[cam];
  const int st       = starts[cam];

  if (tid < 32) {                       // wave 0 only (uniform branch)
    const int nelem = ngt * 4;          // fp32 elements in the GT tile
    if (nelem > 0) {
      const uint64_t ga = (uint64_t)(uintptr_t)(const void*)(gt_boxes + (size_t)st * 4);
      const uint32_t lds_addr = (uint32_t)(uintptr_t)(void*)lds_gtb;
      // D# group 0: count=1, lds_addr, global_addr(57b), type=2
      u32x4 g0;
      g0[0] = 1u;
      g0[1] = lds_addr;
      g0[2] = (uint32_t)(ga & 0xFFFFFFFFull);
      g0[3] = (uint32_t)((ga >> 32) & 0x1FFFFFFull) | (2u << 30);
      // D# group 1: wg_mask=0, data_size=2 (4B), tensor_dim0=nelem,
      // tensor_dim1=1, tile_dim0=nelem (1-D tile), dim0_stride=nelem
      i32x8 g1;
      g1[0] = 0x00020000;
      g1[1] = (int)((nelem & 0xFFFF) << 16);
      g1[2] = (int)(((nelem >> 16) & 0xFFFF) | (1 << 16));
      g1[3] = (int)((nelem & 0xFFFF) << 16);
      g1[4] = 0;
      g1[5] = nelem;
      g1[6] = 0;
      g1[7] = 0;
      i32x4 z4 = {0, 0, 0, 0};
      i32x8 z8 = {0, 0, 0, 0, 0, 0, 0, 0};
      __builtin_amdgcn_tensor_load_to_lds(g0, g1, z4, z4, z8, 0);
    }
    if (tid < ngt) lds_gtl[tid] = gt_labels[st + tid];
    __builtin_amdgcn_s_wait_tensorcnt(0);
  }
  __syncthreads();

  float accPos = 0.f, accNeg = 0.f, accBox = 0.f;
  float accObj = 0.f, accNoobj = 0.f, accCe = 0.f;

  const float4* pb4 = (const float4*)pred_boxes + (size_t)cam * A_;
  const float*  po  = pred_obj    + (size_t)cam * A_;
  const float*  pl  = pred_labels + (size_t)cam * A_ * C_;

  for (int a = blkInCam * TPB + tid; a < A_; a += BLOCKS_PER_CAM * TPB) {
    const float4 p = pb4[a];
    const float phw = p.z * 0.5f, phh = p.w * 0.5f;
    const float p1x = p.x - phw, p1y = p.y - phh;
    const float p2x = p.x + phw, p2y = p.y + phh;
    const float areaP = p.z * p.w;

    float best = -1.0f; int bi = 0;
    for (int k = 0; k < ngt; ++k) {
      const float gx = lds_gtb[k * 4 + 0], gy = lds_gtb[k * 4 + 1];
      const float gw = lds_gtb[k * 4 + 2], gh = lds_gtb[k * 4 + 3];
      const float ghw = gw * 0.5f, ghh = gh * 0.5f;
      float ix = fminf(p2x, gx + ghw) - fmaxf(p1x, gx - ghw);
      float iy = fminf(p2y, gy + ghh) - fmaxf(p1y, gy - ghh);
      ix = fmaxf(ix, 0.0f); iy = fmaxf(iy, 0.0f);
      const float inter = ix * iy;
      const float iou = inter / (areaP + gw * gh - inter + 1e-6f);
      if (iou > best) { best = iou; bi = k; }   // first-max, matches argmax
    }

    const float o = po[a];
    if (best > IOU_THR) {                       // ngt==0 => best==-1 => neg
      accPos += 1.0f;
      const float gx = lds_gtb[bi * 4 + 0], gy = lds_gtb[bi * 4 + 1];
      const float gw = lds_gtb[bi * 4 + 2], gh = lds_gtb[bi * 4 + 3];
      const float dx = p.x - gx, dy = p.y - gy, dw = p.z - gw, dh = p.w - gh;
      accBox += dx * dx + dy * dy + dw * dw + dh * dh;
      accObj += softplus_f(-o);
      // cross-entropy only where pos: skips ~100% of pred_labels traffic
      const float* x = pl + (size_t)a * C_;
      float m = x[0];
      #pragma unroll
      for (int j = 1; j < C_; ++j) m = fmaxf(m, x[j]);
      float s = 0.0f;
      #pragma unroll
      for (int j = 0; j < C_; ++j) s += expf(x[j] - m);
      const float lse = m + logf(s);
      accCe += lse - x[lds_gtl[bi]];
    } else {
      accNeg += 1.0f;
      accNoobj += softplus_f(o);
    }
  }

  // Wave-level reduction on the matrix pipe (all 32 lanes active here).
  const float wPos   = wave_sum_wmma(accPos);
  const float wNeg   = wave_sum_wmma(accNeg);
  const float wBox   = wave_sum_wmma(accBox);
  const float wObj   = wave_sum_wmma(accObj);
  const float wNoobj = wave_sum_wmma(accNoobj);
  const float wCe    = wave_sum_wmma(accCe);

  if ((tid & 31) == 0) {
    atomicAdd(&accs[cam * 6 + 0], wPos);
    atomicAdd(&accs[cam * 6 + 1], wNeg);
    atomicAdd(&accs[cam * 6 + 2], wBox);
    atomicAdd(&accs[cam * 6 + 3], wObj);
    atomicAdd(&accs[cam * 6 + 4], wNoobj);
    atomicAdd(&accs[cam * 6 + 5], wCe);
  }
}

// ---------------------------------------------------------------------------
// Kernel 2: per-camera safe-means + global combination into 5 outputs.
// ---------------------------------------------------------------------------
__global__ void yolo_finalize_kernel(const float* __restrict__ accs,
                                     float* __restrict__ out) {
  __shared__ float sh[CAMS][6];
  const int t = threadIdx.x;
  if (t < CAMS) {
    const float pos   = accs[t * 6 + 0];
    const float neg   = accs[t * 6 + 1];
    const float box   = accs[t * 6 + 2];
    const float obj   = accs[t * 6 + 3];
    const float noobj = accs[t * 6 + 4];
    const float ce    = accs[t * 6 + 5];
    sh[t][0] = LAM_COORD * box;                               // box_loss_c (sum)
    sh[t][1] = (pos > 0.f) ? LAM_OBJ   * (obj   / pos) : 0.f; // obj_loss_c
    sh[t][2] = (neg > 0.f) ? LAM_NOOBJ * (noobj / neg) : 0.f; // noobj_loss_c
    sh[t][3] = (pos > 0.f) ? LAM_CLS   * (ce    / pos) : 0.f; // cls_loss_c
    sh[t][4] = pos;
    sh[t][5] = neg;
  }
  __syncthreads();
  if (t == 0) {
    float sb = 0.f, so = 0.f, sn = 0.f, sc = 0.f, np = 0.f, nn = 0.f;
    for (int i = 0; i < CAMS; ++i) {
      sb += sh[i][0]; so += sh[i][1]; sn += sh[i][2];
      sc += sh[i][3]; np += sh[i][4]; nn += sh[i][5];
    }
    const float box_loss   = (np > 0.f) ? sb / fmaxf(np, 1.f) : 0.f;
    const float obj_loss   = (np > 0.f) ? so / fmaxf(np, 1.f) : 0.f;
    const float cls_loss   = (np > 0.f) ? sc / fmaxf(np, 1.f) : 0.f;
    const float noobj_loss = (nn > 0.f) ? sn / fmaxf(nn, 1.f) : 0.f;
    out[0] = box_loss + obj_loss + noobj_loss + cls_loss;
    out[1] = box_loss;
    out[2] = obj_loss;
    out[3] = noobj_loss;
    out[4] = cls_loss;
  }
}

// ---------------------------------------------------------------------------
extern "C" void kernel_launch(void* const* d_in, const int* in_sizes, int n_in,
                              void* d_out, int out_size, void* d_ws, size_t ws_size,
                              hipStream_t stream) {
  const float* pred_boxes  = (const float*)d_in[0];
  const float* pred_labels = (const float*)d_in[1];
  const float* pred_obj    = (const float*)d_in[2];
  const float* gt_boxes    = (const float*)d_in[3];
  const int*   gt_labels   = (const int*)d_in[4];
  const int*   start_end   = (const int*)d_in[5];
  float* out = (float*)d_out;

  int*   ws_i   = (int*)d_ws;
  int*   starts = ws_i;                               // 48 ints
  int*   ngt    = ws_i + 64;                          // 48 ints
  float* accs   = (float*)((char*)d_ws + 1024);       // 48*6 floats

  yolo_setup_kernel<<<1, 256, 0, stream>>>(start_end, starts, ngt, accs);
  yolo_loss_main<<<CAMS * BLOCKS_PER_CAM, TPB, 0, stream>>>(
      pred_boxes, pred_labels, pred_obj, gt_boxes, gt_labels,
      starts, ngt, accs);
  yolo_finalize_kernel<<<1, 64, 0, stream>>>(accs, out);
}